// ZVectorSparseRouter_489626272104
// MI455X (gfx1250) — compile-verified
//
#include <hip/hip_runtime.h>
#include <math.h>

#define HIDDEN      2048
#define ROUTE_H     256
#define NUM_EXPERTS 64
#define BATCH       32
#define SEQ         4096
#define TOPK        8
#define LN_EPS      1e-5f

#define POOL_CHUNKS    32                       // partial chunks per batch
#define ROWS_PER_CHUNK (SEQ / POOL_CHUNKS)      // 128 rows per block
#define POOL_THREADS   512                      // 512 threads * float4 = 2048 cols

typedef float v2f __attribute__((ext_vector_type(2)));
typedef float v4f __attribute__((ext_vector_type(4)));
typedef float v8f __attribute__((ext_vector_type(8)));

// ---------------------------------------------------------------------------
// Stage 1: streaming partial mean-pool.  grid = BATCH*POOL_CHUNKS blocks.
// Fully coalesced b128 non-temporal loads; each block reduces 128 rows.
// ---------------------------------------------------------------------------
__global__ __launch_bounds__(POOL_THREADS)
void pool_partial_kernel(const float* __restrict__ x, float* __restrict__ partial) {
    const int blk = blockIdx.x;
    const int b   = blk / POOL_CHUNKS;
    const int c   = blk % POOL_CHUNKS;
    const int col4 = threadIdx.x;            // covers columns [4*col4, 4*col4+3]

    const v4f* base = (const v4f*)(x + (size_t)b * SEQ * HIDDEN
                                     + (size_t)c * ROWS_PER_CHUNK * HIDDEN) + col4;
    v4f acc = {0.f, 0.f, 0.f, 0.f};
#pragma unroll 8
    for (int r = 0; r < ROWS_PER_CHUNK; ++r) {
        v4f v = __builtin_nontemporal_load(base + (size_t)r * (HIDDEN / 4));
        acc += v;
    }
    v4f* outp = (v4f*)(partial + ((size_t)b * POOL_CHUNKS + c) * HIDDEN) + col4;
    *outp = acc;
}

// ---------------------------------------------------------------------------
// Stage 2: deterministic reduction of partials -> pooled[B][H] (mean).
// ---------------------------------------------------------------------------
__global__ __launch_bounds__(POOL_THREADS)
void pool_reduce_kernel(const float* __restrict__ partial, float* __restrict__ pooled) {
    const int b    = blockIdx.x;
    const int col4 = threadIdx.x;
    const v4f* base = (const v4f*)(partial + (size_t)b * POOL_CHUNKS * HIDDEN) + col4;
    v4f acc = {0.f, 0.f, 0.f, 0.f};
#pragma unroll
    for (int c = 0; c < POOL_CHUNKS; ++c)
        acc += base[(size_t)c * (HIDDEN / 4)];
    acc *= (1.0f / (float)SEQ);
    ((v4f*)(pooled + (size_t)b * HIDDEN))[col4] = acc;
}

// ---------------------------------------------------------------------------
// Fused router: GEMM1 (WMMA f32 16x16x4) + LayerNorm + erf-GELU +
// GEMM2 (WMMA) + top-8 softmax scatter.  One block, 32 waves (wave32).
// ---------------------------------------------------------------------------
__global__ __launch_bounds__(1024)
void router_kernel(const float* __restrict__ pooled,
                   const float* __restrict__ W1, const float* __restrict__ b1,
                   const float* __restrict__ ln_g, const float* __restrict__ ln_b,
                   const float* __restrict__ W2, const float* __restrict__ b2,
                   float* __restrict__ out) {
    __shared__ float h[BATCH][ROUTE_H];            // 32 KB
    __shared__ float logits[BATCH][NUM_EXPERTS];   // 8 KB

    const int tid  = threadIdx.x;
    const int wave = tid >> 5;
    const int lane = tid & 31;
    const int l    = lane & 15;      // position within 16-lane half
    const int half = lane >> 4;      // 0 or 1

    // ---- GEMM1: h[32,256] = pooled[32,2048] @ W1[2048,256] + b1 ------------
    {
        const int m0 = (wave >> 4) * 16;   // 2 M tiles
        const int n0 = (wave & 15) * 16;   // 16 N tiles
        v8f acc = {0.f, 0.f, 0.f, 0.f, 0.f, 0.f, 0.f, 0.f};
        const float* Arow = pooled + (size_t)(m0 + l) * HIDDEN;
        const float* Bcol = W1 + (n0 + l);
        for (int k = 0; k < HIDDEN; k += 4) {
            // A 16x4 fp32: VGPR0/1 hold K = k+2*half, k+2*half+1 for this lane
            v2f a = *(const v2f*)(Arow + k + 2 * half);
            v2f bb;
            bb.x = Bcol[(size_t)(k + 2 * half) * ROUTE_H];
            bb.y = Bcol[(size_t)(k + 2 * half + 1) * ROUTE_H];
            acc = __builtin_amdgcn_wmma_f32_16x16x4_f32(
                      false, a, false, bb, (short)0, acc, false, false);
        }
        const float bias = b1[n0 + l];
#pragma unroll
        for (int r = 0; r < 8; ++r)
            h[m0 + r + 8 * half][n0 + l] = acc[r] + bias;
    }
    __syncthreads();

    // ---- LayerNorm + exact GELU: wave w handles row w -----------------------
    {
        const int row = wave;
        float v[8];
        float s = 0.f, ss = 0.f;
#pragma unroll
        for (int i = 0; i < 8; ++i) {
            v[i] = h[row][lane * 8 + i];
            s  += v[i];
            ss += v[i] * v[i];
        }
#pragma unroll
        for (int off = 16; off >= 1; off >>= 1) {
            s  += __shfl_xor(s, off, 32);
            ss += __shfl_xor(ss, off, 32);
        }
        const float mu  = s * (1.0f / (float)ROUTE_H);
        const float var = ss * (1.0f / (float)ROUTE_H) - mu * mu;
        const float rs  = rsqrtf(var + LN_EPS);
#pragma unroll
        for (int i = 0; i < 8; ++i) {
            const int c = lane * 8 + i;
            float y = (v[i] - mu) * rs * ln_g[c] + ln_b[c];
            y = 0.5f * y * (1.0f + erff(y * 0.70710678118654752f));
            h[row][c] = y;
        }
    }
    __syncthreads();

    // ---- GEMM2: logits[32,64] = h[32,256] @ W2[256,64] + b2 ----------------
    if (wave < 8) {                       // wave-uniform branch: EXEC stays all-1s
        const int m0 = (wave >> 2) * 16;  // 2 M tiles
        const int n0 = (wave & 3) * 16;   // 4 N tiles
        v8f acc = {0.f, 0.f, 0.f, 0.f, 0.f, 0.f, 0.f, 0.f};
        const float* Bcol = W2 + (n0 + l);
        for (int k = 0; k < ROUTE_H; k += 4) {
            v2f a = *(const v2f*)&h[m0 + l][k + 2 * half];
            v2f bb;
            bb.x = Bcol[(size_t)(k + 2 * half) * NUM_EXPERTS];
            bb.y = Bcol[(size_t)(k + 2 * half + 1) * NUM_EXPERTS];
            acc = __builtin_amdgcn_wmma_f32_16x16x4_f32(
                      false, a, false, bb, (short)0, acc, false, false);
        }
        const float bias = b2[n0 + l];
#pragma unroll
        for (int r = 0; r < 8; ++r)
            logits[m0 + r + 8 * half][n0 + l] = acc[r] + bias;
    }
    __syncthreads();

    // ---- top-8 + softmax + scatter: one thread per batch row ----------------
    if (tid < BATCH) {
        const int row = tid;
        float topv[TOPK];
        int   topi[TOPK];
#pragma unroll
        for (int i = 0; i < TOPK; ++i) {
            float best = -INFINITY;
            int   bi   = 0;
            for (int e = 0; e < NUM_EXPERTS; ++e) {
                const float val = logits[row][e];
                if (val > best) { best = val; bi = e; }   // first-hit tie-break, matches top_k
            }
            topv[i] = best;
            topi[i] = bi;
            logits[row][bi] = -INFINITY;
        }
        const float mx = topv[0];
        float w[TOPK];
        float sum = 0.f;
#pragma unroll
        for (int i = 0; i < TOPK; ++i) { w[i] = expf(topv[i] - mx); sum += w[i]; }
        const float inv = 1.0f / sum;
        float* orow = out + (size_t)row * NUM_EXPERTS;
        for (int e = 0; e < NUM_EXPERTS; ++e) orow[e] = 0.f;
#pragma unroll
        for (int i = 0; i < TOPK; ++i) orow[topi[i]] = w[i] * inv;
    }
}

// ---------------------------------------------------------------------------
extern "C" void kernel_launch(void* const* d_in, const int* in_sizes, int n_in,
                              void* d_out, int out_size, void* d_ws, size_t ws_size,
                              hipStream_t stream) {
    (void)in_sizes; (void)n_in; (void)out_size; (void)ws_size;
    const float* x    = (const float*)d_in[0];
    const float* W1   = (const float*)d_in[1];
    const float* b1   = (const float*)d_in[2];
    const float* ln_g = (const float*)d_in[3];
    const float* ln_b = (const float*)d_in[4];
    const float* W2   = (const float*)d_in[5];
    const float* b2   = (const float*)d_in[6];
    // d_in[7] = top_k scalar (fixed at 8, dims are compile-time)
    float* out = (float*)d_out;

    float* partial = (float*)d_ws;                                        // B*CHUNKS*H
    float* pooled  = partial + (size_t)BATCH * POOL_CHUNKS * HIDDEN;      // B*H

    pool_partial_kernel<<<BATCH * POOL_CHUNKS, POOL_THREADS, 0, stream>>>(x, partial);
    pool_reduce_kernel<<<BATCH, POOL_THREADS, 0, stream>>>(partial, pooled);
    router_kernel<<<1, 1024, 0, stream>>>(pooled, W1, b1, ln_g, ln_b, W2, b2, out);
}